// RoutingCapsules_188978561673
// MI455X (gfx1250) — compile-verified
//
#include <hip/hip_runtime.h>
#include <hip/hip_bf16.h>

// ---------------------------------------------------------------------------
// Capsule dynamic routing, fused-recompute formulation for MI455X (gfx1250).
//
// Shapes: x[B=256, C=1152, I=8] fp32, W[N=10, C=1152, D=16, I=8] fp32,
// out v[B=256, N=10, D=16] fp32.
//
// Routing logits are linear in the accumulated v's:  b = u_hat . V_cum,
// V_cum = sum of previous squash outputs. So each of the 3 iterations is:
//   w = softmax_n(u_hat . V_cum);  s = sum_c w * u_hat;  v = squash(s)
// with u_hat computed on the fly via V_WMMA_F32_16X16X4_F32 pairs
// (16x16x8 fp32 GEMM per (n,c) over a 16-batch tile, transposed so M=d, N=b).
// u_hat is recomputed in the weighting pass (WMMA is the cheapest resource)
// so no 80-VGPR u[] array stays live -> kernel fits well under 256 VGPRs.
//
// Workspace (floats): V_cum[256*10*16] then partials[16][256*10*16] = 2.66 MB.
// Deterministic: no atomics; all cross-wave / cross-chunk sums in fixed order.
// ---------------------------------------------------------------------------

#define B_TOT 256
#define C_TOT 1152
#define I_DIM 8
#define N_CAPS 10
#define D_DIM 16
#define BT 16                      // batch tile (WMMA N dim)
#define NBT (B_TOT / BT)           // 16 batch tiles
#define CCHUNK 16                  // c-chunks (partial slots)
#define C_PER_WG (C_TOT / CCHUNK)  // 72
#define WAVES 4
#define C_PER_WAVE (C_PER_WG / WAVES)     // 18
#define TILE_ELEMS (BT * N_CAPS * D_DIM)  // 2560 floats per (chunk, btile)

typedef float v2f __attribute__((ext_vector_type(2)));
typedef float v8f __attribute__((ext_vector_type(8)));

// ds_swizzle offset encodings (group-of-32: and=offset[4:0], or=[9:5], xor=[14:10])
#define SWZ_XOR(x) (((x) << 10) | 0x1F)

__device__ __forceinline__ float lane_xor_addf(float v, const int xor_mask_enc) {
    int i = __builtin_bit_cast(int, v);
    int o;
    switch (xor_mask_enc) {
        case 1:  o = __builtin_amdgcn_ds_swizzle(i, SWZ_XOR(1));  break;
        case 2:  o = __builtin_amdgcn_ds_swizzle(i, SWZ_XOR(2));  break;
        case 4:  o = __builtin_amdgcn_ds_swizzle(i, SWZ_XOR(4));  break;
        case 8:  o = __builtin_amdgcn_ds_swizzle(i, SWZ_XOR(8));  break;
        default: o = __builtin_amdgcn_ds_swizzle(i, SWZ_XOR(16)); break;
    }
    return v + __builtin_bit_cast(float, o);
}

// ---------------------------------------------------------------------------
// Kernel A: one routing pass. Grid = NBT*CCHUNK blocks of 128 threads (4 waves).
// Each wave: 16 batch x all 10 caps x 18 in-caps, u_hat via f32 WMMA.
// C/D tile layout (M=d, N=b): vgpr r, lane l -> d = r + 8*(l>>4), b = l&15.
// ---------------------------------------------------------------------------
__global__ __launch_bounds__(128) void caps_route(
    const float* __restrict__ x, const float* __restrict__ W,
    const float* __restrict__ Vc, float* __restrict__ part) {
    __shared__ float red[WAVES][TILE_ELEMS];  // 40 KB

    const int tid  = threadIdx.x;
    const int wave = tid >> 5;
    const int lane = tid & 31;
    const int l16  = lane & 15;
    const int hi   = lane >> 4;

    const int btile = blockIdx.x & (NBT - 1);
    const int cblk  = blockIdx.x >> 4;
    const int b0    = btile * BT;
    const int c0    = cblk * C_PER_WG + wave * C_PER_WAVE;

    v8f s[N_CAPS];
#pragma unroll
    for (int n = 0; n < N_CAPS; ++n) s[n] = (v8f)0.0f;

    // per-lane bases
    const float* xbase = x + (size_t)(b0 + l16) * C_TOT * I_DIM + 2 * hi;
    const float* vbase = Vc + (size_t)(b0 + l16) * (N_CAPS * D_DIM) + hi * 8;
    const float* wlane = W + (size_t)l16 * I_DIM + 2 * hi;

    for (int cc = 0; cc < C_PER_WAVE; ++cc) {
        const int c = c0 + cc;
        // B fragments (x^T, K=i, N=b): vgpr0 = K {0|2}, vgpr1 = K {1|3}
        const float* xp = xbase + (size_t)c * I_DIM;
        v2f xf0 = {xp[0], xp[1]};  // i = 2hi, 2hi+1
        v2f xf1 = {xp[4], xp[5]};  // i = 4+2hi, 5+2hi

        // per-c W base; per-n stride (589824 B) folds into the 24-bit IOFFSET
        const float* wq = wlane + (size_t)c * (D_DIM * I_DIM);

        float logit[N_CAPS];
        // ---- pass 1: routing logits (u_hat . V_cum) ----
#pragma unroll
        for (int n = 0; n < N_CAPS; ++n) {
            const float* wp = wq + n * (C_TOT * D_DIM * I_DIM);
            v2f a0 = {wp[0], wp[1]};
            v2f a1 = {wp[4], wp[5]};
            v8f acc = (v8f)0.0f;
            acc = __builtin_amdgcn_wmma_f32_16x16x4_f32(false, a0, false, xf0,
                                                        (short)0, acc, false, false);
            acc = __builtin_amdgcn_wmma_f32_16x16x4_f32(false, a1, false, xf1,
                                                        (short)0, acc, false, false);
            const float* vp = vbase + n * D_DIM;  // loop-invariant: hoisted to regs
            float t = acc[0] * vp[0];
            t = fmaf(acc[1], vp[1], t);
            t = fmaf(acc[2], vp[2], t);
            t = fmaf(acc[3], vp[3], t);
            t = fmaf(acc[4], vp[4], t);
            t = fmaf(acc[5], vp[5], t);
            t = fmaf(acc[6], vp[6], t);
            t = fmaf(acc[7], vp[7], t);
            t = lane_xor_addf(t, 16);  // add the other d-half (lane ^ 16)
            logit[n] = t;
        }
        // softmax over 10 caps; logits bounded ~|0.4| by construction, no max
        // subtraction needed (softmax is shift-invariant).
        float se = 0.0f;
#pragma unroll
        for (int n = 0; n < N_CAPS; ++n) {
            logit[n] = __expf(logit[n]);
            se += logit[n];
        }
        const float inv = 1.0f / se;
        // ---- pass 2: recompute u_hat, accumulate s += w * u_hat ----
        // K-fragments accumulated in REVERSED order: same sum, different
        // expression tree, so CSE cannot resurrect an 80-VGPR u[] array.
#pragma unroll
        for (int n = 0; n < N_CAPS; ++n) {
            const float* wp = wq + n * (C_TOT * D_DIM * I_DIM);
            v2f a0 = {wp[0], wp[1]};
            v2f a1 = {wp[4], wp[5]};
            v8f acc = (v8f)0.0f;
            acc = __builtin_amdgcn_wmma_f32_16x16x4_f32(false, a1, false, xf1,
                                                        (short)0, acc, false, false);
            acc = __builtin_amdgcn_wmma_f32_16x16x4_f32(false, a0, false, xf0,
                                                        (short)0, acc, false, false);
            s[n] += (logit[n] * inv) * acc;
        }
    }

    // dump wave partials to LDS: red[wave][(b_local*N + n)*D + d]
    {
        float* rb = &red[wave][(l16 * N_CAPS) * D_DIM + hi * 8];
#pragma unroll
        for (int n = 0; n < N_CAPS; ++n) {
#pragma unroll
            for (int r = 0; r < 8; ++r) rb[n * D_DIM + r] = s[n][r];
        }
    }
    __syncthreads();

    // fixed-order 4-wave reduction -> per-chunk partial (deterministic, no atomics)
    float* out = part + ((size_t)cblk * B_TOT + b0) * (N_CAPS * D_DIM);
    for (int k = tid; k < TILE_ELEMS; k += 128) {
        float v = ((red[0][k] + red[1][k]) + red[2][k]) + red[3][k];
        out[k] = v;
    }
}

// ---------------------------------------------------------------------------
// Kernel B: sum chunk partials (fixed order), squash, update V_cum or emit out.
// Grid = B blocks of 160 threads (tid -> n = tid/16, d = tid%16).
// ---------------------------------------------------------------------------
__global__ __launch_bounds__(160) void caps_finish(
    const float* __restrict__ part, float* __restrict__ Vc,
    float* __restrict__ out, int final_iter) {
    const int b   = blockIdx.x;
    const int tid = threadIdx.x;
    const size_t idx = (size_t)b * (N_CAPS * D_DIM) + tid;

    float s = 0.0f;
#pragma unroll
    for (int ch = 0; ch < CCHUNK; ++ch)
        s += part[(size_t)ch * (B_TOT * N_CAPS * D_DIM) + idx];

    // sum of squares over the 16 d-lanes of this capsule (xor tree in-wave)
    float sq = s * s;
    sq = lane_xor_addf(sq, 1);
    sq = lane_xor_addf(sq, 2);
    sq = lane_xor_addf(sq, 4);
    sq = lane_xor_addf(sq, 8);

    const float scale = (sq / (1.0f + sq)) / (sqrtf(sq) + 1e-8f);
    const float v = s * scale;
    if (final_iter)
        out[idx] = v;
    else
        Vc[idx] += v;
}

__global__ void caps_zero(float* __restrict__ p, int n) {
    int i = blockIdx.x * blockDim.x + threadIdx.x;
    if (i < n) p[i] = 0.0f;
}

extern "C" void kernel_launch(void* const* d_in, const int* in_sizes, int n_in,
                              void* d_out, int out_size, void* d_ws, size_t ws_size,
                              hipStream_t stream) {
    const float* x = (const float*)d_in[0];  // [256,1152,8]
    const float* W = (const float*)d_in[1];  // [10,1152,16,8]
    float* out = (float*)d_out;              // [256,10,16]

    float* Vc   = (float*)d_ws;                         // 40960 floats
    float* part = Vc + (size_t)B_TOT * N_CAPS * D_DIM;  // 16 * 40960 floats

    const int state = B_TOT * N_CAPS * D_DIM;  // 40960
    caps_zero<<<(state + 255) / 256, 256, 0, stream>>>(Vc, state);

    for (int it = 0; it < 3; ++it) {
        caps_route<<<NBT * CCHUNK, 128, 0, stream>>>(x, W, Vc, part);
        caps_finish<<<B_TOT, 160, 0, stream>>>(part, Vc, out, it == 2 ? 1 : 0);
    }
}